// RiRoIAlignRotated_33260226740674
// MI455X (gfx1250) — compile-verified
//
#include <hip/hip_runtime.h>

typedef __attribute__((ext_vector_type(2))) float v2f;
typedef __attribute__((ext_vector_type(8))) float v8f;

// Problem constants (match reference)
constexpr int   CTOT  = 256;        // total channels = C * N_ORI
constexpr int   Hf    = 256, Wf = 256;
constexpr int   HW    = Hf * Wf;
constexpr int   OH    = 7, OW = 7;
constexpr int   NBIN  = OH * OW;    // 49
constexpr int   NORI  = 8;
constexpr int   CG    = CTOT / NORI;  // 32 channel groups
constexpr int   NPTS  = NBIN * 4;   // 196 sample points (2x2 per bin)
constexpr int   NCOL  = CG * NBIN;  // 1568 mixing columns
constexpr int   NTILE = NCOL / 16;  // 98 WMMA column tiles
constexpr float SCALE = 0.25f;
constexpr float ORI_OVER_2PI = 1.27323954473516f;  // 8/(2*pi)

__device__ __forceinline__ int imin(int a, int b) { return a < b ? a : b; }

__global__ __launch_bounds__(224) void riroi_align_rotated_kernel(
    const float* __restrict__ feat,   // (B, 256, 256, 256)
    const float* __restrict__ rois,   // (n_rois, 6)
    float* __restrict__ out)          // (n_rois, 256, 7, 7)
{
    __shared__ float acc[CTOT * NBIN];   // 50176 bytes: vals[ct][bin] pre-mixing

    const int roi_i = blockIdx.x;
    const float* roi = rois + roi_i * 6;

    // ---- per-ROI scalars (uniform -> scalar loads/ALU) ----
    const int   b     = (int)roi[0];
    const float cx    = roi[1] * SCALE;
    const float cy    = roi[2] * SCALE;
    const float rw    = fmaxf(roi[3] * SCALE, 1.0f);
    const float rh    = fmaxf(roi[4] * SCALE, 1.0f);
    const float theta = roi[5];
    const float bin_h = rh * (1.0f / OH);
    const float bin_w = rw * (1.0f / OW);
    const float ind_f  = theta * ORI_OVER_2PI;
    const float ind_fl = floorf(ind_f);
    const float l_var  = ind_f - ind_fl;
    const float r_var  = 1.0f - l_var;
    int ind = (int)ind_fl;
    ind = ((ind % NORI) + NORI) % NORI;
    float sin_t, cos_t;
    __sincosf(theta, &sin_t, &cos_t);   // CLOCKWISE == false

    const int tid = threadIdx.x;

    // ================= Stage 1: rotated bilinear gather =================
    // thread p < 196 owns one sample point: p = ((oh*7+ow)*2+gy)*2+gx
    const bool active = tid < NPTS;
    int   o00 = 0, o01 = 0, o10 = 0, o11 = 0;
    float w00 = 0.f, w01 = 0.f, w10 = 0.f, w11 = 0.f;
    if (active) {
        const int bin = tid >> 2, s = tid & 3;
        const int oh = bin / OW, ow = bin % OW;
        const int gy = s >> 1, gx = s & 1;
        const float yy = -0.5f * rh + ((float)oh + ((float)gy + 0.5f) * 0.5f) * bin_h;
        const float xx = -0.5f * rw + ((float)ow + ((float)gx + 0.5f) * 0.5f) * bin_w;
        float y = yy * cos_t - xx * sin_t + cy;
        float x = yy * sin_t + xx * cos_t + cx;
        const bool valid = (y >= -1.0f) && (y <= (float)Hf) && (x >= -1.0f) && (x <= (float)Wf);
        y = fmaxf(y, 0.0f);
        x = fmaxf(x, 0.0f);
        const int y0 = imin((int)floorf(y), Hf - 1);
        const int x0 = imin((int)floorf(x), Wf - 1);
        const int y1 = imin(y0 + 1, Hf - 1);
        const int x1 = imin(x0 + 1, Wf - 1);
        const float ly = (y0 >= Hf - 1) ? 0.0f : (y - (float)y0);
        const float lx = (x0 >= Wf - 1) ? 0.0f : (x - (float)x0);
        const float hy = 1.0f - ly, hx = 1.0f - lx;
        const float vs = valid ? 0.25f : 0.0f;   // fold validity + 2x2-mean
        w00 = hy * hx * vs;  w01 = hy * lx * vs;
        w10 = ly * hx * vs;  w11 = ly * lx * vs;
        o00 = y0 * Wf + x0;  o01 = y0 * Wf + x1;
        o10 = y1 * Wf + x0;  o11 = y1 * Wf + x1;
    }

    const float* fb = feat + (size_t)b * CTOT * HW;
    const int  mybin  = tid >> 2;
    const bool writer = active && ((tid & 3) == 0);

#pragma unroll 4
    for (int ct = 0; ct < CTOT; ++ct) {
        const float* f = fb + (size_t)ct * HW;
        float val = 0.0f;
        if (active)
            val = w00 * f[o00] + w01 * f[o01] + w10 * f[o10] + w11 * f[o11];
        // quad (2x2 sample) reduction; quads are lane-aligned within a wave32
        val += __shfl_xor(val, 1, 32);
        val += __shfl_xor(val, 2, 32);
        if (writer)
            acc[ct * NBIN + mybin] = val;   // mean already folded via 0.25
    }
    __syncthreads();

    // ================= Stage 2: orientation mixing as WMMA matmul =========
    // out[o, n] = sum_j M[o][j] * vals[j, n],  n = c*49 + bin (1568 cols)
    // M[o][j] = r_var*[j == (o-ind)%8] + l_var*[j == (o-ind+1)%8]
    // V_WMMA_F32_16X16X4_F32, K=8 split into two accumulating K=4 passes.
    const int wave  = tid >> 5;       // 0..6
    const int lane  = tid & 31;
    const int m     = lane & 15;      // A row / D row group / B,D column
    const int khalf = lane >> 4;      // K-subgroup within fragment
    const int k0    = khalf * 2;

    v2f a0 = {0.f, 0.f}, a1 = {0.f, 0.f};
    if (m < NORI) {
        const int jr = (m - ind) & 7;      // (o - ind) mod 8 (two's complement &7 ok)
        const int jl = (jr + 1) & 7;
        auto mv = [&](int j) -> float {
            return (j == jr ? r_var : 0.0f) + (j == jl ? l_var : 0.0f);
        };
        a0.x = mv(k0);     a0.y = mv(k0 + 1);      // K = 0..3 chunk
        a1.x = mv(k0 + 4); a1.y = mv(k0 + 5);      // K = 4..7 chunk
    }

    const size_t obase = (size_t)roi_i * (CTOT * NBIN);
    for (int t = wave; t < NTILE; t += 7) {        // exactly 14 tiles per wave
        const int n    = t * 16 + m;
        const int c    = n / NBIN;
        const int binn = n - c * NBIN;
        const int base = c * (NORI * NBIN) + k0 * NBIN + binn;
        v2f b0, b1;
        b0.x = acc[base];                // j = k0
        b0.y = acc[base + NBIN];         // j = k0+1
        b1.x = acc[base + 4 * NBIN];     // j = k0+4
        b1.y = acc[base + 5 * NBIN];     // j = k0+5
        v8f d = {};
        d = __builtin_amdgcn_wmma_f32_16x16x4_f32(false, a0, false, b0,
                                                  (short)0, d, false, false);
        d = __builtin_amdgcn_wmma_f32_16x16x4_f32(false, a1, false, b1,
                                                  (short)0, d, false, false);
        if (lane < 16) {                 // rows 0..7 (o = 0..7) live in lanes 0..15
            const size_t ob = obase + (size_t)(c * NORI) * NBIN + binn;
#pragma unroll
            for (int o = 0; o < NORI; ++o)
                out[ob + (size_t)o * NBIN] = d[o];
        }
    }
}

extern "C" void kernel_launch(void* const* d_in, const int* in_sizes, int n_in,
                              void* d_out, int out_size, void* d_ws, size_t ws_size,
                              hipStream_t stream) {
    const float* feat = (const float*)d_in[0];
    const float* rois = (const float*)d_in[1];
    float* out = (float*)d_out;
    const int n_rois = in_sizes[1] / 6;
    riroi_align_rotated_kernel<<<n_rois, 224, 0, stream>>>(feat, rois, out);
}